// WaveletEmbedding_69930657513692
// MI455X (gfx1250) — compile-verified
//
#include <hip/hip_runtime.h>
#include <hip/hip_bf16.h>

typedef __attribute__((ext_vector_type(2))) float v2f;
typedef __attribute__((ext_vector_type(8))) float v8f;

// db4 analysis filters (as in the reference)
__constant__ float c_DEC_LO[8] = {
    -0.010597401784997278f,  0.032883011666982945f,  0.030841381835986965f,
    -0.18703481171888114f,  -0.02798376941698385f,   0.6308807679295904f,
     0.7148465705525415f,    0.23037781330885523f };
__constant__ float c_DEC_HI[8] = {
    -0.23037781330885523f,   0.7148465705525415f,   -0.6308807679295904f,
    -0.02798376941698385f,   0.18703481171888114f,   0.030841381835986965f,
    -0.032883011666982945f, -0.010597401784997278f };

// symmetric-extension index map for length-n signal (single reflection is enough here)
__device__ __forceinline__ int symidx(int u, int n) {
    if (u < 0)  u = -u - 1;
    if (u >= n) u = 2 * n - 1 - u;
    return u;
}

// ---------------------------------------------------------------------------
// Precompute Weff[k=c*8+j][n] = sum_{k28} W[n, c*28+k28] * M[k28][j]
// where M is the fused 2-level DWT (rows ordered aa, ad, dd, da).
// weff stored as [24][128] f32 (12 KB) in d_ws.
// ---------------------------------------------------------------------------
__global__ void weff_precompute_kernel(const float* __restrict__ W,
                                       float* __restrict__ weff) {
    int tid = blockIdx.x * blockDim.x + threadIdx.x;
    if (tid >= 24 * 128) return;
    int n  = tid & 127;
    int kj = tid >> 7;        // 0..23
    int c  = kj >> 3;
    int j  = kj & 7;

    float lo[8], hi[8];       // reversed filters (correlation form)
    #pragma unroll
    for (int i = 0; i < 8; ++i) { lo[i] = c_DEC_LO[7 - i]; hi[i] = c_DEC_HI[7 - i]; }

    // Level 1 (n=8): column j of the 7x8 analysis matrices
    float A1[7], D1[7];
    #pragma unroll
    for (int i = 0; i < 7; ++i) {
        float a = 0.f, d = 0.f;
        #pragma unroll
        for (int jj = 0; jj < 8; ++jj) {
            if (symidx(2 * i + jj - 6, 8) == j) { a += lo[jj]; d += hi[jj]; }
        }
        A1[i] = a; D1[i] = d;
    }
    // Level 2 (n=7): compose -> column j of full 28x8 matrix M
    float Mcol[28];
    #pragma unroll
    for (int i = 0; i < 7; ++i) {
        float aa = 0.f, ad = 0.f, dd = 0.f, da = 0.f;
        #pragma unroll
        for (int jj = 0; jj < 8; ++jj) {
            int s = symidx(2 * i + jj - 6, 7);
            aa += lo[jj] * A1[s];
            ad += hi[jj] * A1[s];
            dd += hi[jj] * D1[s];
            da += lo[jj] * D1[s];
        }
        Mcol[i] = aa; Mcol[7 + i] = ad; Mcol[14 + i] = dd; Mcol[21 + i] = da;
    }
    const float* Wr = W + n * 84 + c * 28;
    float acc = 0.f;
    #pragma unroll
    for (int k = 0; k < 28; ++k) acc += Wr[k] * Mcol[k];
    weff[kj * 128 + n] = acc;
}

// ---------------------------------------------------------------------------
// Main GEMM: (819200 x 24) x (24 x 128) via V_WMMA_F32_16X16X4_F32.
// One block = 16 output rows (16 consecutive patches of one (b,v)).
// Wave w (of 8) computes output columns [16w, 16w+16) with 6 K=4 WMMA steps.
// ---------------------------------------------------------------------------
__global__ __launch_bounds__(256) void wavelet_wmma_kernel(
        const float* __restrict__ x,     // (32, 3, 4096, 25)
        const float* __restrict__ weff,  // (24, 128)
        float* __restrict__ out) {       // (32, 25600, 128)
    constexpr int T = 4096, V = 25, C = 3;
    constexpr int TILES_PER_BV = 64;     // 1024 patches / 16 rows

    const int tile = blockIdx.x;         // 0..51199
    const int bv   = tile / TILES_PER_BV;        // b*25 + v
    const int p0   = (tile % TILES_PER_BV) * 16; // first patch of tile
    const int b    = bv / V;
    const int v    = bv % V;

    const int wave  = threadIdx.x >> 5;  // 0..7 -> N block
    const int lane  = threadIdx.x & 31;
    const int half  = lane >> 4;         // 0: lanes 0-15, 1: lanes 16-31
    const int l16   = lane & 15;
    const int nbase = wave * 16;

    // ---- B operand: 6 chunks of (4 x 16) from weff[k][n] ----
    // lanes 0-15 hold rows K=4q,4q+1 ; lanes 16-31 hold rows K=4q+2,4q+3
    v2f Bt[6];
    #pragma unroll
    for (int q = 0; q < 6; ++q) {
        const float* src = weff + (4 * q + 2 * half) * 128 + nbase + l16;
        Bt[q].x = src[0];
        Bt[q].y = src[128];
    }

    // ---- A operand: 6 chunks of (16 x 4) gathered from x with edge clamp ----
    // row m = l16 (patch p0+m); chunk q covers K = 4q..4q+3, K = c*8 + j
    const float* xb = x + (size_t)b * (C * T * V) + v;
    v2f At[6];
    const int tbase = 4 * (p0 + l16);
    #pragma unroll
    for (int q = 0; q < 6; ++q) {
        const int c = q >> 1;
        const int j = (q & 1) * 4 + 2 * half;
        int t0 = tbase + j;     if (t0 > T - 1) t0 = T - 1;   // edge pad
        int t1 = tbase + j + 1; if (t1 > T - 1) t1 = T - 1;
        const float* xc = xb + (size_t)c * (T * V);
        At[q].x = xc[(size_t)t0 * V];
        At[q].y = xc[(size_t)t1 * V];
    }

    // ---- accumulate: 6 x v_wmma_f32_16x16x4_f32 ----
    v8f acc = {};
    #pragma unroll
    for (int q = 0; q < 6; ++q) {
        acc = __builtin_amdgcn_wmma_f32_16x16x4_f32(
            /*neg_a=*/false, At[q], /*neg_b=*/false, Bt[q],
            /*c_mod=*/(short)0, acc, /*reuse_a=*/false, /*reuse_b=*/false);
    }

    // ---- store D: VGPR r -> row (half*8 + r), col nbase+l16 (streaming NT) ----
    const size_t rowbase = (size_t)bv * 1024 + p0 + half * 8;
    float* outp = out + (rowbase * 128) + nbase + l16;
    #pragma unroll
    for (int r = 0; r < 8; ++r) {
        __builtin_nontemporal_store(acc[r], outp + (size_t)r * 128);
    }
}

extern "C" void kernel_launch(void* const* d_in, const int* in_sizes, int n_in,
                              void* d_out, int out_size, void* d_ws, size_t ws_size,
                              hipStream_t stream) {
    const float* x = (const float*)d_in[0];   // (32,3,4096,25) f32
    const float* W = (const float*)d_in[1];   // (128,84) f32
    float* out  = (float*)d_out;              // (32,25600,128) f32
    float* weff = (float*)d_ws;               // 24*128 f32 = 12 KB scratch

    // Fold 2-level DWT into W  ->  Weff (24 x 128)
    weff_precompute_kernel<<<12, 256, 0, stream>>>(W, weff);

    // 819200 rows / 16 per tile = 51200 blocks, 8 wave32 each
    wavelet_wmma_kernel<<<51200, 256, 0, stream>>>(x, weff, out);
}